// Seq2SeqMulti_37864431682689
// MI455X (gfx1250) — compile-verified
//
#include <hip/hip_runtime.h>
#include <math.h>

typedef __attribute__((ext_vector_type(2))) float v2f;
typedef __attribute__((ext_vector_type(8))) float v8f;

#define B_   128
#define H_   512
#define IN_  256
#define E_   256
#define V_   20000
#define TSRC 64
#define TTRG 50

// =====================================================================
// fp32 WMMA GEMM:  C[M,N] = A[M,K] * W[N,K]^T (+ bias[n]) (+ReLU)
// V_WMMA_F32_16X16X4_F32.  Block = 128 threads (4 waves, wave32).
// Wave tile = 16(M) x 64(N) (4 accumulators); block tile = 16 x 256.
// M multiple of 16 (grid.y = M/16), N multiple of 16, K arbitrary.
// Inner loop is branch-free: load columns are clamped to N-1 (valid,
// possibly duplicated data; results discarded at store), and the K tail
// (only K=257 here) is handled in ONE guarded step after the main loop.
// EXEC is all-ones at every WMMA (uniform branches only).
// =====================================================================
__global__ void gemm_wmma_f32(const float* __restrict__ A, int lda,
                              const float* __restrict__ W, int ldw,
                              float* __restrict__ C, int ldc,
                              int N, int K,
                              const float* __restrict__ bias, int relu)
{
    const int wave = threadIdx.x >> 5;
    const int lane = threadIdx.x & 31;
    const int lr   = lane & 15;       // row (A) / col (B,C) within 16-tile
    const int kh   = lane >> 4;       // 0 -> K pair {0,1}; 1 -> K pair {2,3}
    const int m0   = blockIdx.y << 4;
    const int n0   = (blockIdx.x << 8) + (wave << 6);

    const bool tv0 = (n0      < N);   // wave-uniform store guards
    const bool tv1 = (n0 + 16 < N);
    const bool tv2 = (n0 + 32 < N);
    const bool tv3 = (n0 + 48 < N);

    // clamped load columns -> inner loop needs no guards at all
    const int c0 = min(n0      + lr, N - 1);
    const int c1 = min(n0 + 16 + lr, N - 1);
    const int c2 = min(n0 + 32 + lr, N - 1);
    const int c3 = min(n0 + 48 + lr, N - 1);

    const float* __restrict__ Ar = A + (size_t)(m0 + lr) * lda + (kh << 1);
    const float* __restrict__ P0 = W + (size_t)c0 * ldw + (kh << 1);
    const float* __restrict__ P1 = W + (size_t)c1 * ldw + (kh << 1);
    const float* __restrict__ P2 = W + (size_t)c2 * ldw + (kh << 1);
    const float* __restrict__ P3 = W + (size_t)c3 * ldw + (kh << 1);

    v8f acc0 = {}, acc1 = {}, acc2 = {}, acc3 = {};

    const int K4 = K & ~3;
    int k0 = 0;
    #pragma unroll 2
    for (; k0 < K4; k0 += 4) {
        v2f a, b0, b1, b2, b3;
        a.x  = Ar[k0]; a.y  = Ar[k0 + 1];
        b0.x = P0[k0]; b0.y = P0[k0 + 1];
        b1.x = P1[k0]; b1.y = P1[k0 + 1];
        b2.x = P2[k0]; b2.y = P2[k0 + 1];
        b3.x = P3[k0]; b3.y = P3[k0 + 1];
        acc0 = __builtin_amdgcn_wmma_f32_16x16x4_f32(false, a, false, b0, (short)0, acc0, false, false);
        acc1 = __builtin_amdgcn_wmma_f32_16x16x4_f32(false, a, false, b1, (short)0, acc1, false, false);
        acc2 = __builtin_amdgcn_wmma_f32_16x16x4_f32(false, a, false, b2, (short)0, acc2, false, false);
        acc3 = __builtin_amdgcn_wmma_f32_16x16x4_f32(false, a, false, b3, (short)0, acc3, false, false);
    }
    if (k0 < K) {   // single tail step (K % 4 != 0), executed once
        const int ka = k0 + (kh << 1);
        v2f a = {}, b0 = {}, b1 = {}, b2 = {}, b3 = {};
        if (ka < K) {
            a.x = Ar[k0]; b0.x = P0[k0]; b1.x = P1[k0]; b2.x = P2[k0]; b3.x = P3[k0];
        }
        if (ka + 1 < K) {
            a.y = Ar[k0 + 1]; b0.y = P0[k0 + 1]; b1.y = P1[k0 + 1]; b2.y = P2[k0 + 1]; b3.y = P3[k0 + 1];
        }
        acc0 = __builtin_amdgcn_wmma_f32_16x16x4_f32(false, a, false, b0, (short)0, acc0, false, false);
        acc1 = __builtin_amdgcn_wmma_f32_16x16x4_f32(false, a, false, b1, (short)0, acc1, false, false);
        acc2 = __builtin_amdgcn_wmma_f32_16x16x4_f32(false, a, false, b2, (short)0, acc2, false, false);
        acc3 = __builtin_amdgcn_wmma_f32_16x16x4_f32(false, a, false, b3, (short)0, acc3, false, false);
    }

    // C/D layout: VGPR v -> M = m0 + 8*kh + v,  N = tile_base + lr
    const int rbase = m0 + (kh << 3);
    #pragma unroll
    for (int v = 0; v < 8; ++v) {
        const size_t roff = (size_t)(rbase + v) * ldc;
        if (tv0) { int c = n0      + lr; float x = acc0[v] + (bias ? bias[c] : 0.0f); if (relu) x = fmaxf(x, 0.0f); C[roff + c] = x; }
        if (tv1) { int c = n0 + 16 + lr; float x = acc1[v] + (bias ? bias[c] : 0.0f); if (relu) x = fmaxf(x, 0.0f); C[roff + c] = x; }
        if (tv2) { int c = n0 + 32 + lr; float x = acc2[v] + (bias ? bias[c] : 0.0f); if (relu) x = fmaxf(x, 0.0f); C[roff + c] = x; }
        if (tv3) { int c = n0 + 48 + lr; float x = acc3[v] + (bias ? bias[c] : 0.0f); if (relu) x = fmaxf(x, 0.0f); C[roff + c] = x; }
    }
}

// ---------------- GRU gate fusion: h = (1-z)*n + z*h  (optional seq mask)
__global__ void gru_gate(const float* __restrict__ gi, const float* __restrict__ gh,
                         float* __restrict__ h,
                         const int* __restrict__ src_len, int t, int use_mask)
{
    int idx = blockIdx.x * blockDim.x + threadIdx.x;
    if (idx >= B_ * H_) return;
    int b = idx / H_, j = idx - b * H_;
    const float* gib = gi + (size_t)b * (3 * H_);
    const float* ghb = gh + (size_t)b * (3 * H_);
    float r = 1.0f / (1.0f + expf(-(gib[j]          + ghb[j])));
    float z = 1.0f / (1.0f + expf(-(gib[H_ + j]     + ghb[H_ + j])));
    float n = tanhf(gib[2 * H_ + j] + r * ghb[2 * H_ + j]);
    float hn = (1.0f - z) * n + z * h[idx];
    if (!use_mask || t < src_len[b]) h[idx] = hn;
}

// ---------------- decoder input: x[b] = [emb[eid[b]] (256), rate[b] (1)]
__global__ void build_dec_x(const float* __restrict__ emb,
                            const int* __restrict__ eid,
                            const float* __restrict__ rate,
                            float* __restrict__ x)
{
    int b = blockIdx.x;
    int j = threadIdx.x;                      // 256 threads == E_
    x[(size_t)b * 257 + j] = emb[(size_t)eid[b] * E_ + j];
    if (j == 0) x[(size_t)b * 257 + 256] = rate[b];
}

// ---------------- pr input: x[b] = [emb[eid[b]] (256), h[b] (512)]
__global__ void build_pr_x(const float* __restrict__ emb,
                           const int* __restrict__ eid,
                           const float* __restrict__ h,
                           float* __restrict__ x)
{
    int b = blockIdx.x;
    for (int j = threadIdx.x; j < E_ + H_; j += blockDim.x) {
        float v = (j < E_) ? emb[(size_t)eid[b] * E_ + j]
                           : h[(size_t)b * H_ + (j - E_)];
        x[(size_t)b * (E_ + H_) + j] = v;
    }
}

// ---------------- per-row: lse = max + log(sum exp(x-max)); argmax (first occurrence)
__global__ void row_lse_argmax(const float* __restrict__ logits,
                               float* __restrict__ lse, int* __restrict__ amax)
{
    __shared__ float sm[256];
    __shared__ int   si[256];
    __shared__ float ss[256];
    int b = blockIdx.x;
    const float* row = logits + (size_t)b * V_;

    float m = -INFINITY; int mi = 0;
    for (int i = threadIdx.x; i < V_; i += blockDim.x) {
        float v = row[i];
        if (v > m) { m = v; mi = i; }           // strict > keeps first occurrence
    }
    sm[threadIdx.x] = m; si[threadIdx.x] = mi;
    __syncthreads();
    for (int s = 128; s > 0; s >>= 1) {
        if (threadIdx.x < s) {
            float v2 = sm[threadIdx.x + s]; int i2 = si[threadIdx.x + s];
            if (v2 > sm[threadIdx.x] || (v2 == sm[threadIdx.x] && i2 < si[threadIdx.x])) {
                sm[threadIdx.x] = v2; si[threadIdx.x] = i2;
            }
        }
        __syncthreads();
    }
    float rowmax = sm[0];

    float acc = 0.0f;
    for (int i = threadIdx.x; i < V_; i += blockDim.x)
        acc += expf(row[i] - rowmax);
    ss[threadIdx.x] = acc;
    __syncthreads();
    for (int s = 128; s > 0; s >>= 1) {
        if (threadIdx.x < s) ss[threadIdx.x] += ss[threadIdx.x + s];
        __syncthreads();
    }
    if (threadIdx.x == 0) {
        lse[b]  = rowmax + logf(ss[0]);
        amax[b] = si[0];
    }
}

// ---------------- logp = logits - lse  (in place, in the d_out slice)
__global__ void sub_lse(float* __restrict__ logits, const float* __restrict__ lse)
{
    int b = blockIdx.y;
    int i = blockIdx.x * blockDim.x + threadIdx.x;
    if (i < V_) logits[(size_t)b * V_ + i] -= lse[b];
}

// ---------------- rate head: sigmoid( relu(pr1) . W2 + b2 )   (pr1 pre-ReLU'd)
__global__ void pr_final(const float* __restrict__ pr1,
                         const float* __restrict__ W2,
                         const float* __restrict__ b2,
                         float* __restrict__ rate_out,
                         float* __restrict__ rate_carry)
{
    __shared__ float ss[256];
    int b = blockIdx.x;
    float acc = 0.0f;
    for (int j = threadIdx.x; j < H_; j += blockDim.x)
        acc += pr1[(size_t)b * H_ + j] * W2[j];
    ss[threadIdx.x] = acc;
    __syncthreads();
    for (int s = 128; s > 0; s >>= 1) {
        if (threadIdx.x < s) ss[threadIdx.x] += ss[threadIdx.x + s];
        __syncthreads();
    }
    if (threadIdx.x == 0) {
        float p = 1.0f / (1.0f + expf(-(ss[0] + b2[0])));
        rate_out[b]   = p;
        rate_carry[b] = p;
    }
}

__global__ void zero_f32(float* __restrict__ p, long long n)
{
    long long i = (long long)blockIdx.x * blockDim.x + threadIdx.x;
    long long stride = (long long)gridDim.x * blockDim.x;
    for (; i < n; i += stride) p[i] = 0.0f;
}

// =====================================================================
extern "C" void kernel_launch(void* const* d_in, const int* in_sizes, int n_in,
                              void* d_out, int out_size, void* d_ws, size_t ws_size,
                              hipStream_t stream)
{
    const float* src      = (const float*)d_in[0];
    const int*   src_len  = (const int*)  d_in[1];
    const int*   trg_eid  = (const int*)  d_in[2];
    const float* trg_rate = (const float*)d_in[3];
    const float* emb      = (const float*)d_in[4];
    const float* enc_Wi   = (const float*)d_in[5];
    const float* enc_Wh   = (const float*)d_in[6];
    const float* enc_bi   = (const float*)d_in[7];
    const float* enc_bh   = (const float*)d_in[8];
    const float* dec_Wi   = (const float*)d_in[9];
    const float* dec_Wh   = (const float*)d_in[10];
    const float* dec_bi   = (const float*)d_in[11];
    const float* dec_bh   = (const float*)d_in[12];
    const float* We       = (const float*)d_in[13];
    const float* be       = (const float*)d_in[14];
    const float* W1       = (const float*)d_in[15];
    const float* b1       = (const float*)d_in[16];
    const float* W2       = (const float*)d_in[17];
    const float* b2       = (const float*)d_in[18];

    float* eid_out  = (float*)d_out;                         // [50,128,20000]
    float* rate_out = eid_out + (size_t)TTRG * B_ * V_;      // [50,128]

    float* ws   = (float*)d_ws;
    float* h    = ws; ws += B_ * H_;
    float* gi   = ws; ws += B_ * 3 * H_;
    float* gh   = ws; ws += B_ * 3 * H_;
    float* xdec = ws; ws += B_ * 257;
    float* prx  = ws; ws += B_ * (E_ + H_);
    float* pr1  = ws; ws += B_ * H_;
    float* lse  = ws; ws += B_;
    float* ratec= ws; ws += B_;
    int*   eidc = (int*)ws;

    // init: h = 0; output timestep 0 = zeros
    zero_f32<<<64, 256, 0, stream>>>(h, (long long)B_ * H_);
    zero_f32<<<1024, 256, 0, stream>>>(eid_out, (long long)B_ * V_);
    zero_f32<<<1, 128, 0, stream>>>(rate_out, B_);

    const dim3 blk(128);
    const dim3 grid_gates((3 * H_ + 255) / 256, B_ / 16);   // N=1536
    const dim3 grid_logit((V_ + 255) / 256, B_ / 16);       // N=20000
    const dim3 grid_pr1((H_ + 255) / 256, B_ / 16);         // N=512

    // ---------------- encoder: 64 masked GRU steps
    for (int t = 0; t < TSRC; ++t) {
        gemm_wmma_f32<<<grid_gates, blk, 0, stream>>>(
            src + (size_t)t * B_ * IN_, IN_, enc_Wi, IN_, gi, 3 * H_,
            3 * H_, IN_, enc_bi, 0);
        gemm_wmma_f32<<<grid_gates, blk, 0, stream>>>(
            h, H_, enc_Wh, H_, gh, 3 * H_,
            3 * H_, H_, enc_bh, 0);
        gru_gate<<<(B_ * H_ + 255) / 256, 256, 0, stream>>>(gi, gh, h, src_len, t, 1);
    }

    // ---------------- decoder: 49 greedy autoregressive steps
    const int*   eid_ptr  = trg_eid;    // trg_eid[0]  (first B entries)
    const float* rate_ptr = trg_rate;   // trg_rate[0]
    for (int t = 1; t < TTRG; ++t) {
        build_dec_x<<<B_, 256, 0, stream>>>(emb, eid_ptr, rate_ptr, xdec);
        gemm_wmma_f32<<<grid_gates, blk, 0, stream>>>(
            xdec, 257, dec_Wi, 257, gi, 3 * H_, 3 * H_, 257, dec_bi, 0);
        gemm_wmma_f32<<<grid_gates, blk, 0, stream>>>(
            h, H_, dec_Wh, H_, gh, 3 * H_, 3 * H_, H_, dec_bh, 0);
        gru_gate<<<(B_ * H_ + 255) / 256, 256, 0, stream>>>(gi, gh, h, nullptr, 0, 0);

        float* logits = eid_out + (size_t)t * B_ * V_;
        gemm_wmma_f32<<<grid_logit, blk, 0, stream>>>(
            h, H_, We, H_, logits, V_, V_, H_, be, 0);
        row_lse_argmax<<<B_, 256, 0, stream>>>(logits, lse, eidc);
        sub_lse<<<dim3((V_ + 255) / 256, B_), 256, 0, stream>>>(logits, lse);

        build_pr_x<<<B_, 256, 0, stream>>>(emb, eidc, h, prx);
        gemm_wmma_f32<<<grid_pr1, blk, 0, stream>>>(
            prx, E_ + H_, W1, E_ + H_, pr1, H_, H_, E_ + H_, b1, 1);
        pr_final<<<B_, 256, 0, stream>>>(pr1, W2, b2, rate_out + (size_t)t * B_, ratec);

        eid_ptr  = eidc;
        rate_ptr = ratec;
    }
}